// TaskAlignedAssigner_36017595744363
// MI455X (gfx1250) — compile-verified
//
#include <hip/hip_runtime.h>
#include <math.h>

// ---- Problem constants (from setup_inputs) ----
#define M_ANCH 33600          // 160^2 + 80^2 + 40^2 anchors
#define N_GT   400
#define TOPKK  13
#define NUM_C  80
#define M_TILES (M_ANCH/16)   // 2100 (exact)
#define N_TILES (N_GT/16)     // 25   (exact)

typedef float v2f __attribute__((ext_vector_type(2)));
typedef float v8f __attribute__((ext_vector_type(8)));

#define CEPS 1e-7f
#define FOUR_OVER_PI2 0.40528473456935108577f

// =====================================================================
// K1: per-pair matrices. One wave per 16(gt) x 16(anchor) tile.
// bbox_scores tile = OneHot(16x80) * pd_scores^T(80x16) via 20x
// V_WMMA_F32_16X16X4_F32 (exact in f32: one product term per output).
// Then each lane evaluates CIoU for its 8 accumulator elements.
// =====================================================================
__global__ __launch_bounds__(256) void taa_k1_pair(
    const float* __restrict__ pd_scores, const float* __restrict__ pd_bboxes,
    const float* __restrict__ anchor_point, const float* __restrict__ stride_tensor,
    const int* __restrict__ gt_labels, const float* __restrict__ gt_bboxes,
    float* __restrict__ ovl, float* __restrict__ alg, float* __restrict__ mtr)
{
  const int lane = threadIdx.x;                 // 0..31 (wave32)
  const int mt   = blockIdx.x * 8 + threadIdx.y; // m-tile (wave-uniform guard)
  if (mt >= M_TILES) return;
  const int nt = blockIdx.y;
  const int m0 = mt * 16, n0 = nt * 16;
  const int r  = lane & 15;                      // row/col within half-wave
  const int hi = lane >> 4;

  const int   mcol  = m0 + r;                    // this lane's anchor column
  const int   lbl   = gt_labels[n0 + r];         // A-matrix row = lane&15
  const float* psrow = pd_scores + (size_t)mcol * NUM_C;

  v8f acc = {0.f,0.f,0.f,0.f,0.f,0.f,0.f,0.f};
#if __has_builtin(__builtin_amdgcn_wmma_f32_16x16x4_f32)
  // A 16x4 f32 layout: lane l row=l&15; v0 holds K=2*hi, v1 holds K=2*hi+1.
  // B 4x16 f32 layout mirrors A (split halves): lane l col=l&15; same K slots.
  #pragma unroll
  for (int c = 0; c < NUM_C / 4; ++c) {
    const int kb = 4 * c + 2 * hi;
    v2f A; A.x = (lbl == kb)     ? 1.f : 0.f;
           A.y = (lbl == kb + 1) ? 1.f : 0.f;
    v2f B; B.x = psrow[kb];
           B.y = psrow[kb + 1];
    acc = __builtin_amdgcn_wmma_f32_16x16x4_f32(
        /*neg_a=*/false, A, /*neg_b=*/false, B,
        /*c_mod=*/(short)0, acc, /*reuse_a=*/false, /*reuse_b=*/false);
  }
#else
  // Exact fallback: direct gather (identical math).
  #pragma unroll
  for (int v = 0; v < 8; ++v)
    acc[v] = psrow[gt_labels[n0 + v + 8 * hi]];
#endif

  // Per-lane anchor-side data (same anchor for all 8 rows).
  const float4 pb = ((const float4*)pd_bboxes)[mcol];
  const float px1 = pb.x, py1 = pb.y, px2 = pb.z, py2 = pb.w;
  const float st  = stride_tensor[mcol];
  const float xx  = anchor_point[mcol * 2]     * st;
  const float yy  = anchor_point[mcol * 2 + 1] * st;
  const float w2 = px2 - px1, h2 = py2 - py1;
  const float at2 = atanf(w2 / (h2 + CEPS));

  #pragma unroll
  for (int v = 0; v < 8; ++v) {
    const int g = n0 + v + 8 * hi;               // C/D layout: row = v + 8*hi
    const float4 gb = ((const float4*)gt_bboxes)[g];
    const float gx1 = gb.x, gy1 = gb.y, gx2 = gb.z, gy2 = gb.w;
    // ---- CIoU (matches reference ordering) ----
    float iw = fminf(gx2, px2) - fmaxf(gx1, px1); iw = fmaxf(iw, 0.f);
    float ih = fminf(gy2, py2) - fmaxf(gy1, py1); ih = fmaxf(ih, 0.f);
    const float inter = iw * ih;
    const float w1 = gx2 - gx1, h1 = gy2 - gy1;
    const float uni = w1 * h1 + w2 * h2 - inter + CEPS;
    const float iou = inter / uni;
    const float cw = fmaxf(gx2, px2) - fminf(gx1, px1);
    const float ch = fmaxf(gy2, py2) - fminf(gy1, py1);
    const float c2 = cw * cw + ch * ch + CEPS;
    const float dx = px1 + px2 - gx1 - gx2, dy = py1 + py2 - gy1 - gy2;
    const float rho2 = (dx * dx + dy * dy) * 0.25f;
    const float dv = at2 - atanf(w1 / (h1 + CEPS));
    const float vv = FOUR_OVER_PI2 * dv * dv;
    const float alpha = vv / (vv - iou + 1.f + CEPS);
    const float ov = fmaxf(iou - (rho2 / c2 + vv * alpha), 0.f);
    const float o2 = ov * ov;
    const float align = acc[v] * (o2 * o2 * o2);   // score^1 * overlap^6
    const float md = fminf(fminf(xx - gx1, yy - gy1), fminf(gx2 - xx, gy2 - yy));
    const float met = (md > 1e-9f) ? align : 0.f;  // metrics = align * mask_in_gts
    const size_t idx = (size_t)g * M_ANCH + mcol;
    ovl[idx] = ov; alg[idx] = align; mtr[idx] = met;
  }
}

// =====================================================================
// K2: top-13 per GT row via 13 destructive block-argmax rounds.
// Tie-break: lowest index (matches jax.lax.top_k). Destroys mtr.
// =====================================================================
__global__ __launch_bounds__(256) void taa_k2_topk(float* __restrict__ mtr,
                                                   int* __restrict__ topk)
{
  const int n = blockIdx.x, tid = threadIdx.x;
  __shared__ float sv[256];
  __shared__ int   si[256];
  float* row = mtr + (size_t)n * M_ANCH;
  for (int rr = 0; rr < TOPKK; ++rr) {
    float bv = -2.f; int bi = 0;
    for (int m = tid; m < M_ANCH; m += 256) {
      const float v = row[m];
      if (v > bv) { bv = v; bi = m; }            // ascending -> first max kept
    }
    sv[tid] = bv; si[tid] = bi;
    __syncthreads();
    for (int s = 128; s > 0; s >>= 1) {
      if (tid < s) {
        const float v2 = sv[tid + s]; const int i2 = si[tid + s];
        if (v2 > sv[tid] || (v2 == sv[tid] && i2 < si[tid])) { sv[tid] = v2; si[tid] = i2; }
      }
      __syncthreads();
    }
    if (tid == 0) { topk[n * TOPKK + rr] = si[0]; row[si[0]] = -1.f; }
    __syncthreads();
  }
}

// ---- K0: zero claim counters / per-gt maxima ----
__global__ void taa_k0_init(int* __restrict__ cnt, unsigned* __restrict__ posA,
                            unsigned* __restrict__ posO)
{
  const int t = blockIdx.x * 256 + threadIdx.x;
  if (t < M_ANCH) cnt[t] = 0;
  if (t < N_GT) { posA[t] = 0u; posO[t] = 0u; }
}

// ---- K3a: scatter top-k claims (ANDed with in_gts) ----
__global__ void taa_k3_scatter(const int* __restrict__ topk,
                               const float* __restrict__ ap, const float* __restrict__ st,
                               const float* __restrict__ gtb,
                               int* __restrict__ cnt, int* __restrict__ g1)
{
  const int t = blockIdx.x * 256 + threadIdx.x;
  if (t >= N_GT * TOPKK) return;
  const int n = t / TOPKK;
  const int m = topk[t];
  const float s = st[m];
  const float xx = ap[m * 2] * s, yy = ap[m * 2 + 1] * s;
  const float4 gb = ((const float4*)gtb)[n];
  const float md = fminf(fminf(xx - gb.x, yy - gb.y), fminf(gb.z - xx, gb.w - yy));
  if (md > 1e-9f) { atomicAdd(&cnt[m], 1); g1[m] = n; }   // g1 valid iff cnt==1
}

// ---- K3b: per-anchor resolve; per-gt maxima via monotone atomicMax ----
__global__ __launch_bounds__(256) void taa_k3_anchor(
    const float* __restrict__ ovl, const float* __restrict__ alg,
    const int* __restrict__ cnt, const int* __restrict__ g1,
    int* __restrict__ tgt, int* __restrict__ fgm,
    unsigned* __restrict__ posA, unsigned* __restrict__ posO)
{
  const int m = blockIdx.x * 256 + threadIdx.x;
  if (m >= M_ANCH) return;
  // column argmax of overlaps over ALL gts (first-max tie-break = jnp.argmax)
  float best = ovl[m]; int bi = 0;
  for (int n = 1; n < N_GT; ++n) {
    if (n + 8 < N_GT)
      __builtin_prefetch(&ovl[(size_t)(n + 8) * M_ANCH + m], 0, 1); // global_prefetch_b8
    const float v = ovl[(size_t)n * M_ANCH + m];
    if (v > best) { best = v; bi = n; }
  }
  const int c  = cnt[m];
  const int fg = (c > 0) ? 1 : 0;
  const int g  = (c == 0) ? 0 : ((c == 1) ? g1[m] : bi);
  tgt[m] = g; fgm[m] = fg;
  if (fg) {
    const size_t idx = (size_t)g * M_ANCH + m;
    atomicMax(&posA[g], __float_as_uint(alg[idx]));  // values >= 0: bit-order == float-order
    atomicMax(&posO[g], __float_as_uint(ovl[idx]));
  }
}

// ---- K4: per-anchor outputs (labels, boxes, fg, gt_idx, norm) ----
__global__ __launch_bounds__(256) void taa_k4_final(
    const float* __restrict__ alg, const int* __restrict__ gt_labels,
    const float* __restrict__ gtb, const int* __restrict__ tgt,
    const int* __restrict__ fgm, const unsigned* __restrict__ posA,
    const unsigned* __restrict__ posO, float* __restrict__ norm,
    float* __restrict__ out)
{
  const int m = blockIdx.x * 256 + threadIdx.x;
  if (m >= M_ANCH) return;
  const int g = tgt[m], fg = fgm[m];
  out[m] = (float)gt_labels[g];                              // target_labels
  ((float4*)(out + M_ANCH))[m] = ((const float4*)gtb)[g];    // target_bboxes
  float nv = 0.f;
  if (fg) {
    const float pa = __uint_as_float(posA[g]);
    const float po = __uint_as_float(posO[g]);
    nv = alg[(size_t)g * M_ANCH + m] * po / (pa + 1e-9f);
  }
  norm[m] = nv;
  out[(size_t)85 * M_ANCH + m] = fg ? 1.f : 0.f;             // fg_mask
  out[(size_t)86 * M_ANCH + m] = (float)g;                   // target_gt_idx
}

// ---- K5: scaled one-hot scores, coalesced along class dim ----
__global__ __launch_bounds__(256) void taa_k5_scores(
    const int* __restrict__ tgt, const int* __restrict__ fgm,
    const int* __restrict__ gt_labels, const float* __restrict__ norm,
    float* __restrict__ outS)
{
  const int t = blockIdx.x * 256 + threadIdx.x;
  if (t >= M_ANCH * NUM_C) return;
  const int m = t / NUM_C, c = t % NUM_C;
  const int fg = fgm[m];
  const int lbl = gt_labels[tgt[m]];
  outS[t] = (fg && c == lbl) ? norm[m] : 0.f;
}

// =====================================================================
extern "C" void kernel_launch(void* const* d_in, const int* in_sizes, int n_in,
                              void* d_out, int out_size, void* d_ws, size_t ws_size,
                              hipStream_t stream)
{
  const float* pd_scores     = (const float*)d_in[0];
  const float* pd_bboxes     = (const float*)d_in[1];
  const float* anchor_point  = (const float*)d_in[2];
  const float* stride_tensor = (const float*)d_in[3];
  const int*   gt_labels     = (const int*)  d_in[4];
  const float* gt_bboxes     = (const float*)d_in[5];
  float* out = (float*)d_out;

  // Workspace layout (~162 MB): three N*M fp32 matrices + small arrays.
  const size_t NM = (size_t)N_GT * M_ANCH;
  float*    ovl  = (float*)d_ws;
  float*    alg  = ovl + NM;
  float*    mtr  = alg + NM;
  float*    norm = mtr + NM;
  int*      topk = (int*)(norm + M_ANCH);
  int*      cnt  = topk + N_GT * TOPKK;
  int*      g1   = cnt + M_ANCH;
  int*      tgt  = g1 + M_ANCH;
  int*      fgm  = tgt + M_ANCH;
  unsigned* posA = (unsigned*)(fgm + M_ANCH);
  unsigned* posO = posA + N_GT;

  const int b256M = (M_ANCH + 255) / 256;        // 132

  taa_k1_pair<<<dim3((M_TILES + 7) / 8, N_TILES), dim3(32, 8), 0, stream>>>(
      pd_scores, pd_bboxes, anchor_point, stride_tensor, gt_labels, gt_bboxes,
      ovl, alg, mtr);
  taa_k0_init<<<b256M, 256, 0, stream>>>(cnt, posA, posO);
  taa_k2_topk<<<N_GT, 256, 0, stream>>>(mtr, topk);
  taa_k3_scatter<<<(N_GT * TOPKK + 255) / 256, 256, 0, stream>>>(
      topk, anchor_point, stride_tensor, gt_bboxes, cnt, g1);
  taa_k3_anchor<<<b256M, 256, 0, stream>>>(ovl, alg, cnt, g1, tgt, fgm, posA, posO);
  taa_k4_final<<<b256M, 256, 0, stream>>>(alg, gt_labels, gt_bboxes, tgt, fgm,
                                          posA, posO, norm, out);
  taa_k5_scores<<<(M_ANCH * NUM_C + 255) / 256, 256, 0, stream>>>(
      tgt, fgm, gt_labels, norm, out + (size_t)5 * M_ANCH);
}